// BoundingBoxPrompter_352187318715
// MI455X (gfx1250) — compile-verified
//
#include <hip/hip_runtime.h>

typedef __attribute__((ext_vector_type(2))) float v2f;
typedef __attribute__((ext_vector_type(8))) float v8f;

#define B_    16
#define H_    64
#define W_    64
#define C_    768
#define PH_   32
#define PW_   32
#define NBOX  6
#define HWC_  (H_ * W_ * C_)      /* 3,145,728 elements per batch image */
#define GROUPS 6                  /* chunk-groups per pixel               */
#define CPW    8                  /* chunks (of 16 ch) per wave -> 128 ch */

// One wave handles one pixel x 128 contiguous channels (8 WMMA tiles).
// Per tile: D = A(16x4, all 0.25) x B(4x16, prompt bcast) + C(x tile)
//             = x + prompt        (exact in f32, 4 * 0.25 * p == p).
__global__ __launch_bounds__(256) void bbox_prompt_add_wmma(
    const float* __restrict__ x,    // [16,64,64,768]
    const float* __restrict__ y,    // [6,4]
    const float* __restrict__ bp,   // [32,32,768]
    float* __restrict__ out)        // [16,64,64,768]
{
    const int tid  = threadIdx.x;
    const int lane = tid & 31;
    const int gwid = blockIdx.x * 8 + (tid >> 5);   // global wave id
    const int p    = gwid / GROUPS;                 // pixel index 0..4095
    const int g    = gwid % GROUPS;                 // channel group 0..5
    const int h    = p >> 6;
    const int w    = p & (W_ - 1);
    const int n    = lane & 15;                     // N column for this lane
    const int m0   = (lane >> 4) * 8;               // C/D layout: hi lanes M+8

    // ---------------- first-covering-box selection (once per wave) ----------
    const float sx = (float)W_ / 1024.0f;           // 0.0625
    const float sy = (float)H_ / 1024.0f;           // 0.0625
    int win_xm = 0, win_ym = 0, win_xM = 0, win_yM = 0;
    int found = 0;
#pragma unroll
    for (int i = 0; i < NBOX; ++i) {
        const float b0 = y[i * 4 + 0], b1 = y[i * 4 + 1];
        const float b2 = y[i * 4 + 2], b3 = y[i * 4 + 3];
        const int valid = (b0 >= 0.f) && (b1 >= 0.f) && (b2 >= 0.f) && (b3 >= 0.f);
        const int x1g = (int)fminf(fmaxf(floorf(b0 * sx), 0.f), (float)(W_ - 1));
        const int y1g = (int)fminf(fmaxf(floorf(b1 * sy), 0.f), (float)(H_ - 1));
        const int x2g = (int)fminf(fmaxf(floorf(b2 * sx), 0.f), (float)(W_ - 1));
        const int y2g = (int)fminf(fmaxf(floorf(b3 * sy), 0.f), (float)(H_ - 1));
        const int xm = min(x1g, x2g), xM = max(x1g, x2g);
        const int ym = min(y1g, y2g), yM = max(y1g, y2g);
        const int cov  = valid && (h >= ym) && (h <= yM) && (w >= xm) && (w <= xM);
        const int take = cov && !found;              // first cover wins
        win_xm = take ? xm : win_xm;
        win_xM = take ? xM : win_xM;
        win_ym = take ? ym : win_ym;
        win_yM = take ? yM : win_yM;
        found  = found || cov;
    }

    // ---------------- bilinear setup (align_corners=False), once per wave ---
    const float bh = (float)(win_yM - win_ym + 1);
    const float bw = (float)(win_xM - win_xm + 1);
    const float srcy = fmaxf(((float)(h - win_ym) + 0.5f) * ((float)PH_ / bh) - 0.5f, 0.f);
    const float srcx = fmaxf(((float)(w - win_xm) + 0.5f) * ((float)PW_ / bw) - 0.5f, 0.f);
    int y0i = (int)floorf(srcy);
    int x0i = (int)floorf(srcx);
    y0i = min(y0i, PH_ - 1);                 // no-op when found; safety when not
    x0i = min(x0i, PW_ - 1);
    const int y1i = min(y0i + 1, PH_ - 1);
    const int x1i = min(x0i + 1, PW_ - 1);
    float fy = srcy - (float)y0i;
    float fx = srcx - (float)x0i;
    float gy = 1.f - fy, gx = 1.f - fx;
    const float fm = found ? 1.f : 0.f;      // zero the prompt outside boxes
    fy *= fm;  gy *= fm;                     // pr = gy*(..) + fy*(..) -> 0

    // tap row bases (element offsets into bp), shared by all 8 chunks
    const int t00 = (y0i * PW_ + x0i) * C_;
    const int t01 = (y0i * PW_ + x1i) * C_;
    const int t10 = (y1i * PW_ + x0i) * C_;
    const int t11 = (y1i * PW_ + x1i) * C_;
    const float* bpc = bp + g * (CPW * 16) + n;

    // x/out lane bases; per-chunk delta is 64B -> folds into inst immediates
    const float* xb = x   + (size_t)p * C_ + g * (CPW * 16) + n + (size_t)m0 * HWC_;
    float*       ob = out + (size_t)p * C_ + g * (CPW * 16) + n + (size_t)m0 * HWC_;

    v2f A;  A.x = 0.25f; A.y = 0.25f;        // A[m,k] = 0.25 for all m,k

#pragma unroll
    for (int k = 0; k < CPW; ++k) {
        const int co = k * 16;               // chunk channel offset (elements)

        const float v00 = bpc[t00 + co];
        const float v01 = bpc[t01 + co];
        const float v10 = bpc[t10 + co];
        const float v11 = bpc[t11 + co];
        const float pr = gy * (gx * v00 + fx * v01)
                       + fy * (gx * v10 + fx * v11);
        v2f Bv; Bv.x = pr; Bv.y = pr;        // B[k,n] = prompt[c] for all k

        v8f Cv;
#pragma unroll
        for (int r = 0; r < 8; ++r)
            Cv[r] = __builtin_nontemporal_load(xb + (size_t)r * HWC_ + co);

        v8f D = __builtin_amdgcn_wmma_f32_16x16x4_f32(
            /*neg_a=*/false, A, /*neg_b=*/false, Bv,
            /*c_mod=*/(short)0, Cv, /*reuse_a=*/false, /*reuse_b=*/false);

#pragma unroll
        for (int r = 0; r < 8; ++r)
            __builtin_nontemporal_store(D[r], ob + (size_t)r * HWC_ + co);
    }
}

extern "C" void kernel_launch(void* const* d_in, const int* in_sizes, int n_in,
                              void* d_out, int out_size, void* d_ws, size_t ws_size,
                              hipStream_t stream) {
    (void)in_sizes; (void)n_in; (void)out_size; (void)d_ws; (void)ws_size;
    const float* x  = (const float*)d_in[0];
    const float* y  = (const float*)d_in[1];
    const float* bp = (const float*)d_in[2];
    float* out = (float*)d_out;

    // 4096 pixels * 6 groups = 24576 waves; 8 waves per 256-thread block
    const int blocks = (H_ * W_ * GROUPS) / 8;   // 3072
    bbox_prompt_add_wmma<<<blocks, 256, 0, stream>>>(x, y, bp, out);
}